// TemporalTransformer_38268158608026
// MI455X (gfx1250) — compile-verified
//
#include <hip/hip_runtime.h>
#include <math.h>

// ---- problem constants (from reference) ----
#define B_   8
#define N_   307
#define T_   48
#define D_   64
#define H_   4
#define HD_  16
#define FE_  4
#define SK_  20            // SAMPLE_K = min(48, 5*ceil(log(48))) = 20
#define R_   (B_ * N_ * T_)   // 117888 token rows
#define EPS_ 1e-5f

typedef __attribute__((ext_vector_type(2))) float v2f;
typedef __attribute__((ext_vector_type(8))) float v8f;

#define WMMA_F32(A, Bv, C) \
  __builtin_amdgcn_wmma_f32_16x16x4_f32(false, (A), false, (Bv), (short)0, (C), false, false)

// -------------------------------------------------------------------------
// Kernel 1: time bias  bias[b,t,:] = pe[t,:] + emb_min[..]+emb_hr[..]+...
// -------------------------------------------------------------------------
__global__ void timebias_kernel(const int* __restrict__ tf,
                                const float* __restrict__ emin,
                                const float* __restrict__ ehr,
                                const float* __restrict__ ewd,
                                const float* __restrict__ emo,
                                const float* __restrict__ eyr,
                                float* __restrict__ biasbt) {
  int bt = blockIdx.x;          // b*T + t
  int d  = threadIdx.x;         // 0..63
  int t  = bt % T_;
  const int* f = tf + bt * 5;
  int   i2   = d & ~1;
  float freq = __expf(-logf(10000.0f) * (float)i2 / (float)D_);
  float ang  = (float)t * freq;
  float pe   = (d & 1) ? cosf(ang) : sinf(ang);
  float v = pe + emin[f[0] * D_ + d] + ehr[f[1] * D_ + d] + ewd[f[2] * D_ + d]
               + emo[f[3] * D_ + d] + eyr[f[4] * D_ + d];
  biasbt[bt * D_ + d] = v;
}

// -------------------------------------------------------------------------
// Kernel 2: enc = x + bias[b,t]   (float4 vectorized)
// -------------------------------------------------------------------------
__global__ void enc_kernel(const float4* __restrict__ x,
                           const float4* __restrict__ biasbt,
                           float4* __restrict__ enc, int total4) {
  int i = blockIdx.x * blockDim.x + threadIdx.x;
  if (i >= total4) return;
  int row = i >> 4;                 // 16 float4 per 64-wide row
  int dq  = i & 15;
  int t   = row % T_;
  int b   = row / (N_ * T_);
  float4 xx = x[i];
  float4 bb = biasbt[(b * T_ + t) * 16 + dq];
  float4 r;
  r.x = xx.x + bb.x; r.y = xx.y + bb.y; r.z = xx.z + bb.z; r.w = xx.w + bb.w;
  enc[i] = r;
}

// -------------------------------------------------------------------------
// Kernel 3: fp32 WMMA GEMM, Nc = 64:  Y[R,64] = X[R,Kd] @ W[Kd,64] (+bias)
// One wave computes a full 16x64 output strip: one A fragment feeds 4 WMMAs
// per K-step (4x less X traffic than tile-per-wave).
// -------------------------------------------------------------------------
__global__ void gemm64_wmma_kernel(const float* __restrict__ X,
                                   const float* __restrict__ W,
                                   const float* __restrict__ bias,
                                   float* __restrict__ Y,
                                   int R, int Kd) {
  int wave = (blockIdx.x * blockDim.x + threadIdx.x) >> 5;
  if (wave >= (R >> 4)) return;               // wave-uniform: EXEC stays all-1
  int lane = threadIdx.x & 31;
  int half = lane >> 4;                        // K-pair selector
  int lm   = lane & 15;                        // M (A) / N (B,C,D) index
  int r0 = wave << 4;

  v8f a0 = {0.f,0.f,0.f,0.f,0.f,0.f,0.f,0.f};
  v8f a1 = a0, a2 = a0, a3 = a0;
  const float* xrow = X + (size_t)(r0 + lm) * Kd;
  for (int k0 = 0; k0 < Kd; k0 += 4) {
    int k = k0 + 2 * half;
    v2f A = { xrow[k], xrow[k + 1] };          // A[M=lm][k], A[M=lm][k+1]
    const float* w0 = W + (size_t)k * D_ + lm;
    const float* w1 = W + (size_t)(k + 1) * D_ + lm;
    v2f B0 = { w0[0],  w1[0]  };
    v2f B1 = { w0[16], w1[16] };
    v2f B2 = { w0[32], w1[32] };
    v2f B3 = { w0[48], w1[48] };
    a0 = WMMA_F32(A, B0, a0);
    a1 = WMMA_F32(A, B1, a1);
    a2 = WMMA_F32(A, B2, a2);
    a3 = WMMA_F32(A, B3, a3);
  }
  float bb0 = 0.f, bb1 = 0.f, bb2 = 0.f, bb3 = 0.f;
  if (bias) {
    bb0 = bias[lm]; bb1 = bias[lm + 16]; bb2 = bias[lm + 32]; bb3 = bias[lm + 48];
  }
#pragma unroll
  for (int j = 0; j < 8; ++j) {
    float* yr = Y + (size_t)(r0 + j + 8 * half) * D_ + lm;   // D: M=j+8*half, N=lm
    yr[0]  = a0[j] + bb0;
    yr[16] = a1[j] + bb1;
    yr[32] = a2[j] + bb2;
    yr[48] = a3[j] + bb3;
  }
}

// -------------------------------------------------------------------------
// Kernel 4: ProbSparse attention, one block (64 thr) per (b,h,n).
// Writes ctx in place over the Q buffer (each block owns its head slice).
// -------------------------------------------------------------------------
__global__ void attn_kernel(const float* __restrict__ Qg,
                            const float* __restrict__ Kg,
                            const float* __restrict__ Vg,
                            const int* __restrict__ samp,
                            float* __restrict__ ctx) {
  __shared__ float sQ[T_][HD_], sK[T_][HD_], sV[T_][HD_];
  __shared__ float sM[T_];
  __shared__ int   sel[SK_];
  __shared__ int   pos[T_];
  __shared__ float sUpd[SK_][HD_];
  __shared__ float vmean[HD_];

  int tid = threadIdx.x;
  int bhn = blockIdx.x;
  int n = bhn % N_;
  int h = (bhn / N_) % H_;
  int b = bhn / (N_ * H_);
  size_t base = ((size_t)(b * N_ + n)) * T_ * D_ + h * HD_;

  for (int i = tid; i < T_ * HD_; i += 64) {
    int t = i >> 4, d = i & 15;
    sQ[t][d] = Qg[base + (size_t)t * D_ + d];
    sK[t][d] = Kg[base + (size_t)t * D_ + d];
    sV[t][d] = Vg[base + (size_t)t * D_ + d];
  }
  __syncthreads();

  // sparsity measure M[t] = max_s(q_t . k_samp) - mean_s(q_t . k_samp)
  if (tid < T_) {
    float mx = -3.4e38f, sm = 0.f;
    for (int s = 0; s < SK_; ++s) {
      int idx = samp[tid * SK_ + s];
      float dot = 0.f;
#pragma unroll
      for (int d = 0; d < HD_; ++d) dot += sQ[tid][d] * sK[idx][d];
      mx = fmaxf(mx, dot);
      sm += dot;
    }
    sM[tid]  = mx - sm * (1.0f / SK_);
    pos[tid] = -1;
  }
  __syncthreads();

  // exact top-SK_ via rank (stable: ties broken by lower index, like lax.top_k)
  if (tid < T_) {
    float mv = sM[tid];
    int rank = 0;
    for (int j = 0; j < T_; ++j) {
      float o = sM[j];
      if (o > mv || (o == mv && j < tid)) ++rank;
    }
    if (rank < SK_) sel[rank] = tid;
  }
  __syncthreads();

  if (tid < SK_) pos[sel[tid]] = tid;
  if (tid < HD_) {
    float s = 0.f;
    for (int t = 0; t < T_; ++t) s += sV[t][tid];
    vmean[tid] = s * (1.0f / T_);
  }
  __syncthreads();

  // full attention for the selected queries (2-pass softmax)
  if (tid < SK_) {
    int qr = sel[tid];
    float mx = -3.4e38f;
    for (int t = 0; t < T_; ++t) {
      float dot = 0.f;
#pragma unroll
      for (int d = 0; d < HD_; ++d) dot += sQ[qr][d] * sK[t][d];
      mx = fmaxf(mx, dot * 0.25f);            // 1/sqrt(HD) = 0.25
    }
    float acc[HD_];
#pragma unroll
    for (int d = 0; d < HD_; ++d) acc[d] = 0.f;
    float den = 0.f;
    for (int t = 0; t < T_; ++t) {
      float dot = 0.f;
#pragma unroll
      for (int d = 0; d < HD_; ++d) dot += sQ[qr][d] * sK[t][d];
      float e = __expf(dot * 0.25f - mx);
      den += e;
#pragma unroll
      for (int d = 0; d < HD_; ++d) acc[d] += e * sV[t][d];
    }
    float inv = 1.0f / den;
#pragma unroll
    for (int d = 0; d < HD_; ++d) sUpd[tid][d] = acc[d] * inv;
  }
  __syncthreads();

  // ctx: mean-V everywhere, scatter updates at selected rows
  for (int i = tid; i < T_ * HD_; i += 64) {
    int t = i >> 4, d = i & 15;
    int u = pos[t];
    ctx[base + (size_t)t * D_ + d] = (u >= 0) ? sUpd[u][d] : vmean[d];
  }
}

// -------------------------------------------------------------------------
// Kernel 5: h = LayerNorm(a + b) * g + be ; one wave per 64-wide row
// -------------------------------------------------------------------------
__device__ __forceinline__ float wave_sum32(float v) {
#pragma unroll
  for (int off = 16; off > 0; off >>= 1) v += __shfl_xor(v, off, 32);
  return v;
}

__global__ void add_ln_kernel(const float* __restrict__ a,
                              const float* __restrict__ b,
                              const float* __restrict__ g,
                              const float* __restrict__ be,
                              float* __restrict__ y, int R) {
  int wave = (blockIdx.x * blockDim.x + threadIdx.x) >> 5;
  if (wave >= R) return;
  int lane = threadIdx.x & 31;
  size_t off = (size_t)wave * D_;
  float x0 = a[off + lane]      + b[off + lane];
  float x1 = a[off + lane + 32] + b[off + lane + 32];
  float mean = wave_sum32(x0 + x1) * (1.0f / D_);
  float d0 = x0 - mean, d1 = x1 - mean;
  float var = wave_sum32(d0 * d0 + d1 * d1) * (1.0f / D_);
  float rs = rsqrtf(var + EPS_);
  y[off + lane]      = d0 * rs * g[lane]      + be[lane];
  y[off + lane + 32] = d1 * rs * g[lane + 32] + be[lane + 32];
}

// -------------------------------------------------------------------------
// Kernel 6: fused FFN + residual + LayerNorm.
// One wave per 16 rows. h tile staged in LDS (reused by GEMM1 A-fragments and
// the residual), mid = relu(h@W1+b1) in LDS, ff = mid@W2+b2, out = LN(ff+h).
// A fragments shared across 4 column tiles per WMMA group.
// -------------------------------------------------------------------------
__global__ void ffn_ln_kernel(const float* __restrict__ h,
                              const float* __restrict__ W1,
                              const float* __restrict__ b1,
                              const float* __restrict__ W2,
                              const float* __restrict__ b2,
                              const float* __restrict__ g2,
                              const float* __restrict__ be2,
                              float* __restrict__ out) {
  __shared__ float sh[16][D_];          // 16 x 64 h tile
  __shared__ float mid[16][FE_ * D_];   // 16 x 256
  __shared__ float ftile[16][D_];       // 16 x 64
  int lane = threadIdx.x & 31;
  int half = lane >> 4;
  int lm   = lane & 15;
  int r0   = blockIdx.x << 4;

  // stage h tile
  for (int i = lane; i < 16 * D_; i += 32) {
    int t = i >> 6, d = i & 63;
    sh[t][d] = h[(size_t)(r0 + t) * D_ + d];
  }
  __syncthreads();

  // GEMM1: mid = relu(sh @ W1 + b1); 4 groups of 4 column tiles
  for (int g = 0; g < 4; ++g) {
    int n0 = g * 64;
    v8f a0 = {0.f,0.f,0.f,0.f,0.f,0.f,0.f,0.f};
    v8f a1 = a0, a2 = a0, a3 = a0;
    for (int k0 = 0; k0 < D_; k0 += 4) {
      int k = k0 + 2 * half;
      v2f A = { sh[lm][k], sh[lm][k + 1] };
      const float* w0 = W1 + (size_t)k * (FE_ * D_) + n0 + lm;
      const float* w1 = W1 + (size_t)(k + 1) * (FE_ * D_) + n0 + lm;
      v2f B0 = { w0[0],  w1[0]  };
      v2f B1 = { w0[16], w1[16] };
      v2f B2 = { w0[32], w1[32] };
      v2f B3 = { w0[48], w1[48] };
      a0 = WMMA_F32(A, B0, a0);
      a1 = WMMA_F32(A, B1, a1);
      a2 = WMMA_F32(A, B2, a2);
      a3 = WMMA_F32(A, B3, a3);
    }
    float bb0 = b1[n0 + lm], bb1 = b1[n0 + 16 + lm];
    float bb2 = b1[n0 + 32 + lm], bb3 = b1[n0 + 48 + lm];
#pragma unroll
    for (int j = 0; j < 8; ++j) {
      int row = j + 8 * half;
      float v0 = a0[j] + bb0, v1 = a1[j] + bb1;
      float v2 = a2[j] + bb2, v3 = a3[j] + bb3;
      mid[row][n0 + lm]      = v0 > 0.f ? v0 : 0.f;
      mid[row][n0 + 16 + lm] = v1 > 0.f ? v1 : 0.f;
      mid[row][n0 + 32 + lm] = v2 > 0.f ? v2 : 0.f;
      mid[row][n0 + 48 + lm] = v3 > 0.f ? v3 : 0.f;
    }
  }
  __syncthreads();

  // GEMM2: ftile = mid @ W2 + b2 (one group of 4 column tiles over Nc=64)
  {
    v8f c0 = {0.f,0.f,0.f,0.f,0.f,0.f,0.f,0.f};
    v8f c1 = c0, c2 = c0, c3 = c0;
    for (int k0 = 0; k0 < FE_ * D_; k0 += 4) {
      int k = k0 + 2 * half;
      v2f A = { mid[lm][k], mid[lm][k + 1] };
      const float* w0 = W2 + (size_t)k * D_ + lm;
      const float* w1 = W2 + (size_t)(k + 1) * D_ + lm;
      v2f B0 = { w0[0],  w1[0]  };
      v2f B1 = { w0[16], w1[16] };
      v2f B2 = { w0[32], w1[32] };
      v2f B3 = { w0[48], w1[48] };
      c0 = WMMA_F32(A, B0, c0);
      c1 = WMMA_F32(A, B1, c1);
      c2 = WMMA_F32(A, B2, c2);
      c3 = WMMA_F32(A, B3, c3);
    }
    float bb0 = b2[lm], bb1 = b2[lm + 16], bb2 = b2[lm + 32], bb3 = b2[lm + 48];
#pragma unroll
    for (int j = 0; j < 8; ++j) {
      int row = j + 8 * half;
      ftile[row][lm]      = c0[j] + bb0;
      ftile[row][lm + 16] = c1[j] + bb1;
      ftile[row][lm + 32] = c2[j] + bb2;
      ftile[row][lm + 48] = c3[j] + bb3;
    }
  }
  __syncthreads();

  // residual + LayerNorm: lanes 0..15 each own one row (h from LDS)
  if (lane < 16) {
    int r = r0 + lane;
    float s = 0.f;
    for (int d = 0; d < D_; ++d) {
      float x = ftile[lane][d] + sh[lane][d];
      ftile[lane][d] = x;           // own row: no conflict
      s += x;
    }
    float mean = s * (1.0f / D_);
    float vs = 0.f;
    for (int d = 0; d < D_; ++d) {
      float dd = ftile[lane][d] - mean;
      vs += dd * dd;
    }
    float rs = rsqrtf(vs * (1.0f / D_) + EPS_);
    float* orow = out + (size_t)r * D_;
    for (int d = 0; d < D_; ++d)
      orow[d] = (ftile[lane][d] - mean) * rs * g2[d] + be2[d];
  }
}

// -------------------------------------------------------------------------
// Host launch
// -------------------------------------------------------------------------
extern "C" void kernel_launch(void* const* d_in, const int* in_sizes, int n_in,
                              void* d_out, int out_size, void* d_ws, size_t ws_size,
                              hipStream_t stream) {
  const float* x    = (const float*)d_in[0];
  const int*   tf   = (const int*)d_in[1];
  const int*   samp = (const int*)d_in[2];
  const float* Wq   = (const float*)d_in[3];
  const float* Wk   = (const float*)d_in[4];
  const float* Wv   = (const float*)d_in[5];
  const float* Wo   = (const float*)d_in[6];
  const float* bo   = (const float*)d_in[7];
  const float* W1   = (const float*)d_in[8];
  const float* b1   = (const float*)d_in[9];
  const float* W2   = (const float*)d_in[10];
  const float* b2   = (const float*)d_in[11];
  const float* g1   = (const float*)d_in[12];
  const float* be1  = (const float*)d_in[13];
  const float* g2   = (const float*)d_in[14];
  const float* be2  = (const float*)d_in[15];
  const float* emin = (const float*)d_in[16];
  const float* ehr  = (const float*)d_in[17];
  const float* ewd  = (const float*)d_in[18];
  const float* emo  = (const float*)d_in[19];
  const float* eyr  = (const float*)d_in[20];

  const size_t R64 = (size_t)R_ * D_;
  float* ws     = (float*)d_ws;
  float* biasbt = ws;                       // B*T*64
  float* enc    = biasbt + (size_t)B_ * T_ * D_;
  float* Qb     = enc + R64;                // later reused as ctx
  float* Kb     = Qb + R64;                 // later reused as att_out
  float* Vb     = Kb + R64;                 // later reused as h

  // 1. time/positional bias
  timebias_kernel<<<B_ * T_, D_, 0, stream>>>(tf, emin, ehr, ewd, emo, eyr, biasbt);

  // 2. enc = x + bias
  int total4 = (int)(R64 / 4);
  enc_kernel<<<(total4 + 255) / 256, 256, 0, stream>>>(
      (const float4*)x, (const float4*)biasbt, (float4*)enc, total4);

  // 3. Q/K/V projections (WMMA f32, 16x64 strip per wave)
  int nstrips = R_ / 16;
  int gblocks = (nstrips + 7) / 8;          // 8 waves / 256-thread block
  gemm64_wmma_kernel<<<gblocks, 256, 0, stream>>>(enc, Wq, nullptr, Qb, R_, D_);
  gemm64_wmma_kernel<<<gblocks, 256, 0, stream>>>(enc, Wk, nullptr, Kb, R_, D_);
  gemm64_wmma_kernel<<<gblocks, 256, 0, stream>>>(enc, Wv, nullptr, Vb, R_, D_);

  // 4. ProbSparse attention; ctx overwrites Q buffer
  attn_kernel<<<B_ * H_ * N_, 64, 0, stream>>>(Qb, Kb, Vb, samp, Qb);

  // 5. output projection: att_out = ctx @ Wo + bo  (into K buffer)
  gemm64_wmma_kernel<<<gblocks, 256, 0, stream>>>(Qb, Wo, bo, Kb, R_, D_);

  // 6. h = LN(att_out + enc)  (into V buffer)
  add_ln_kernel<<<(R_ + 7) / 8, 256, 0, stream>>>(Kb, enc, g1, be1, Vb, R_);

  // 7. fused FFN + residual + LN -> d_out
  ffn_ln_kernel<<<R_ / 16, 32, 0, stream>>>(Vb, W1, b1, W2, b2, g2, be2,
                                            (float*)d_out);
}